// LightGCN_54580444398201
// MI455X (gfx1250) — compile-verified
//
#include <hip/hip_runtime.h>

#define N_NODES 100000
#define N_EDGES 1600000
#define DIM     64

typedef __attribute__((ext_vector_type(2))) float v2f;
typedef __attribute__((ext_vector_type(8))) float v8f;

// ---------------- zero deg ----------------
__global__ void k_zero_u32(unsigned* __restrict__ p, int n) {
    int i = blockIdx.x * blockDim.x + threadIdx.x;
    if (i < n) p[i] = 0u;
}

// ---------------- degree histogram over dst ----------------
__global__ void k_deg(const int* __restrict__ dst, unsigned* __restrict__ deg, int nE) {
    for (int e = blockIdx.x * blockDim.x + threadIdx.x; e < nE;
         e += gridDim.x * blockDim.x)
        atomicAdd(&deg[dst[e]], 1u);
}

// ---------------- single-block exclusive scan -> row_ptr, cursor ----------------
__global__ void k_scan(const unsigned* __restrict__ deg, unsigned* __restrict__ row_ptr,
                       unsigned* __restrict__ cursor, int n) {
    __shared__ unsigned sm[1024];
    __shared__ unsigned s_base;
    if (threadIdx.x == 0) { s_base = 0u; row_ptr[0] = 0u; }
    __syncthreads();
    int nchunks = (n + 1023) / 1024;
    for (int c = 0; c < nchunks; ++c) {
        int i = c * 1024 + (int)threadIdx.x;
        unsigned v = (i < n) ? deg[i] : 0u;
        sm[threadIdx.x] = v;
        __syncthreads();
        // Hillis-Steele inclusive scan
        for (int off = 1; off < 1024; off <<= 1) {
            unsigned t = (threadIdx.x >= (unsigned)off) ? sm[threadIdx.x - off] : 0u;
            __syncthreads();
            sm[threadIdx.x] += t;
            __syncthreads();
        }
        unsigned incl = sm[threadIdx.x];
        unsigned base = s_base;
        if (i < n) {
            row_ptr[i + 1] = base + incl;       // exclusive end
            cursor[i]      = base + incl - v;   // exclusive begin
        }
        __syncthreads();
        if (threadIdx.x == 1023) s_base = base + sm[1023];
        __syncthreads();
    }
}

// ---------------- dinv = deg>0 ? rsqrt(deg) : 0 ----------------
__global__ void k_dinv(const unsigned* __restrict__ deg, float* __restrict__ dinv, int n) {
    int i = blockIdx.x * blockDim.x + threadIdx.x;
    if (i < n) {
        unsigned d = deg[i];
        dinv[i] = d ? rsqrtf((float)d) : 0.f;
    }
}

// ---------------- CSR fill: col[] grouped by dst ----------------
__global__ void k_fill(const int* __restrict__ src, const int* __restrict__ dst,
                       unsigned* __restrict__ cursor, unsigned* __restrict__ col, int nE) {
    for (int e = blockIdx.x * blockDim.x + threadIdx.x; e < nE;
         e += gridDim.x * blockDim.x) {
        unsigned p = atomicAdd(&cursor[dst[e]], 1u);
        col[p] = (unsigned)src[e];
    }
}

// ---------------- LGConv propagation: xout[v] = dinv[v] * sum_u dinv[u]*xin[row(u)] ----------------
// One dst node per 16-lane half-wave; D=64 -> float4 per lane. No atomics, all L2 hits.
__global__ void k_prop(const float* __restrict__ xin, const int* __restrict__ ids,
                       const unsigned* __restrict__ row_ptr, const unsigned* __restrict__ col,
                       const float* __restrict__ dinv, float* __restrict__ xout) {
    const int node   = blockIdx.x * 16 + ((int)threadIdx.x >> 4);
    const int lane16 = (int)threadIdx.x & 15;
    const unsigned beg = row_ptr[node];
    const unsigned end = row_ptr[node + 1];
    float4 acc = make_float4(0.f, 0.f, 0.f, 0.f);
    for (unsigned j = beg; j < end; ++j) {
        int s  = (int)col[j];
        float ns = dinv[s];
        int r  = ids ? ids[s] : s;
        // prefetch a future neighbor's feature row into cache (global_prefetch_b8)
        if (j + 4 < end) {
            int sp = (int)col[j + 4];
            int rp = ids ? ids[sp] : sp;
            __builtin_prefetch(xin + (size_t)rp * DIM, 0, 3);
        }
        const float4* xr = (const float4*)(xin + (size_t)r * DIM);
        float4 v = xr[lane16];
        acc.x = fmaf(ns, v.x, acc.x);
        acc.y = fmaf(ns, v.y, acc.y);
        acc.z = fmaf(ns, v.z, acc.z);
        acc.w = fmaf(ns, v.w, acc.w);
    }
    const float dv = dinv[node];
    acc.x *= dv; acc.y *= dv; acc.z *= dv; acc.w *= dv;
    ((float4*)(xout + (size_t)node * DIM))[lane16] = acc;
}

// ---------------- combine: out = w0*emb[ids] + w1*x1 + w2*x2 via V_WMMA_F32_16X16X4_F32 ----------------
// Per 16x16 tile: out_tile = sum_k sum_{jb=0..3} A(=xk[:,4jb:4jb+4]) x B(=w_k * selector)
// A layout (ISA 7.12.2, 32-bit A 16x4): lanes 0-15 -> K=0(v0),K=1(v1); lanes 16-31 -> K=2(v0),K=3(v1).
// B (4x16) mirrors the same K-per-lane-half split; N = lane%16.
// f32 MACs -> numerically equivalent to scalar FMA combine.
__global__ void k_combine(const float* __restrict__ emb, const int* __restrict__ ids,
                          const float* __restrict__ x1, const float* __restrict__ x2,
                          const float* __restrict__ w, float* __restrict__ out) {
    const int wave = blockIdx.x * 8 + ((int)threadIdx.x >> 5);
    const int l    = (int)threadIdx.x & 31;
    const int tr   = wave >> 2;       // 0..6249 row tiles
    const int tc   = wave & 3;        // 0..3 col tiles (D=64)
    const int m    = l & 15;
    const int hi   = l >> 4;          // lane half
    const int gr   = tr * 16 + m;     // global row this lane supplies for A

    const float wk0 = w[0], wk1 = w[1], wk2 = w[2];
    const float* rows0 = emb + (size_t)ids[gr] * DIM;
    const float* rows1 = x1  + (size_t)gr * DIM;
    const float* rows2 = x2  + (size_t)gr * DIM;
    const float* rp[3] = { rows0, rows1, rows2 };
    const float  wk[3] = { wk0, wk1, wk2 };

    const int colbase = tc * 16;
    const int kk = hi * 2;            // K indices this lane holds: kk, kk+1

    v8f c = {};
    #pragma unroll
    for (int k = 0; k < 3; ++k) {
        const float* r = rp[k];
        #pragma unroll
        for (int jb = 0; jb < 4; ++jb) {
            v2f a, b;
            a.x = r[colbase + jb * 4 + kk];
            a.y = r[colbase + jb * 4 + kk + 1];
            b.x = (m == jb * 4 + kk)     ? wk[k] : 0.f;
            b.y = (m == jb * 4 + kk + 1) ? wk[k] : 0.f;
            c = __builtin_amdgcn_wmma_f32_16x16x4_f32(
                    false, a, false, b, (short)0, c, false, false);
        }
    }
    // D layout: VGPR i -> row (i + 8*hi), col m
    float* op = out + (size_t)(tr * 16) * DIM + colbase + m;
    #pragma unroll
    for (int i = 0; i < 8; ++i)
        op[(size_t)(i + 8 * hi) * DIM] = c[i];
}

extern "C" void kernel_launch(void* const* d_in, const int* in_sizes, int n_in,
                              void* d_out, int out_size, void* d_ws, size_t ws_size,
                              hipStream_t stream) {
    const float* emb  = (const float*)d_in[0];
    const int*   ids  = (const int*)d_in[1];   // node_ids (int32 on device)
    const int*   eidx = (const int*)d_in[2];   // edge_index [2, E] int32
    const float* w    = (const float*)d_in[3]; // layer_weights [3]
    float*       out  = (float*)d_out;

    const int* srcA = eidx;            // row 0
    const int* dstA = eidx + N_EDGES;  // row 1

    char*  ws  = (char*)d_ws;
    size_t off = 0;
    auto carve = [&](size_t bytes) -> void* {
        void* p = ws + off;
        off = (off + bytes + 255) & ~(size_t)255;
        return p;
    };
    unsigned* deg    = (unsigned*)carve((size_t)N_NODES * 4);
    unsigned* rowp   = (unsigned*)carve((size_t)(N_NODES + 1) * 4);
    unsigned* cursor = (unsigned*)carve((size_t)N_NODES * 4);
    float*    dinv   = (float*)carve((size_t)N_NODES * 4);
    unsigned* col    = (unsigned*)carve((size_t)N_EDGES * 4);
    float*    x1     = (float*)carve((size_t)N_NODES * DIM * 4);
    float*    x2     = (float*)carve((size_t)N_NODES * DIM * 4);

    k_zero_u32<<<(N_NODES + 255) / 256, 256, 0, stream>>>(deg, N_NODES);
    k_deg<<<1024, 256, 0, stream>>>(dstA, deg, N_EDGES);
    k_scan<<<1, 1024, 0, stream>>>(deg, rowp, cursor, N_NODES);
    k_dinv<<<(N_NODES + 255) / 256, 256, 0, stream>>>(deg, dinv, N_NODES);
    k_fill<<<1024, 256, 0, stream>>>(srcA, dstA, cursor, col, N_EDGES);

    // layer 1: x1 = conv(emb[node_ids]); layer 2: x2 = conv(x1)
    k_prop<<<N_NODES / 16, 256, 0, stream>>>(emb, ids, rowp, col, dinv, x1);
    k_prop<<<N_NODES / 16, 256, 0, stream>>>(x1, nullptr, rowp, col, dinv, x2);

    // out = w0*x0 + w1*x1 + w2*x2  (WMMA f32 16x16x4)
    k_combine<<<(N_NODES / 16) * 4 / 8, 256, 0, stream>>>(emb, ids, x1, x2, w, out);
}